// RecursiveNN_50311246905374
// MI455X (gfx1250) — compile-verified
//
#include <hip/hip_runtime.h>

// RecursiveNN on MI455X (gfx1250), V_WMMA_F32_16X16X4_F32 throughout.
// Leaves = emb[word_ids] @ W1 + b1; then 10 levels of concat(pairs) @ W2 + b2.
//
// Round-3 changes:
//  * 128-leaf subtree per workgroup (512 thr / 16 waves, 129KB dynamic LDS):
//    kernel-1 tree levels have M = 64/32/16 -> no 16x16-tile padding waste.
//  * LDS node stride padded 256 -> 258 floats: A-fragment ds_load_b64 across
//    the 16-lane half-wave hits 32 distinct banks (was 16-way conflict).
//  * W1/W2 pre-packed in per-lane B-fragment order: every B fragment is one
//    global_load_b64 at base + immediate offset.

typedef float v2f __attribute__((ext_vector_type(2)));
typedef float v8f __attribute__((ext_vector_type(8)));

#define VOCAB   50257
#define WDIM    300
#define HID     256
#define BATCH   256
#define LEAVES  1024
#define SUBL    128                 // leaves per workgroup (depth-7 subtree)
#define NODES1  (LEAVES / 8)        // 128 nodes per tree left after kernel 1
#define KS1     (WDIM / 4)          // 75 k-steps, leaf GEMM
#define KS2     ((2 * HID) / 4)     // 128 k-steps, tree GEMM
#define NSTR    258                 // padded LDS node stride (floats)
#define NWAVES  16
#define BLOCK   512
#define LDS_BYTES (SUBL * NSTR * 4) // 132096 B dynamic LDS

__device__ __forceinline__ v8f wmma_f32(v2f a, v2f b, v8f c) {
    // 8 args: (neg_a, A, neg_b, B, c_mod, C, reuse_a, reuse_b)
    return __builtin_amdgcn_wmma_f32_16x16x4_f32(
        false, a, false, b, (short)0, c, false, false);
}

// Pack W[K][HID] into B-fragment order: Wp[((ks*16 + nt)*32 + lane)*2 + v]
//   = W[(ks*4 + (lane>>4)*2 + v) * HID + nt*16 + (lane&15)]
__global__ __launch_bounds__(256) void pack_weights(
    const float* __restrict__ W, float* __restrict__ Wp, int ksteps) {
    const int i = blockIdx.x * 256 + threadIdx.x;
    if (i >= ksteps * 1024) return;
    const int v    = i & 1;
    const int lane = (i >> 1) & 31;
    const int nt   = (i >> 6) & 15;
    const int ks   = i >> 10;
    Wp[i] = W[(ks * 4 + (lane >> 4) * 2 + v) * HID + nt * 16 + (lane & 15)];
}

// One GEMM unit: 16 output rows (M-tile mt) x 32 cols (N-tiles ntp, ntp+1) of
// pairs @ W2 + b2.  A row m = nodes (2m, 2m+1) in the padded LDS buffer;
// the k-loop splits at ks=64 where K crosses from the left to right sibling.
__device__ __forceinline__ void tree_unit(const float* __restrict__ h,
                                          const v2f*   __restrict__ w2,  // +lane
                                          const float* __restrict__ b2,
                                          int mt, int ntp, int l16, int lhalf,
                                          v8f acc[2]) {
#pragma unroll
    for (int j = 0; j < 2; ++j) {
        const float bv = b2[(ntp + j) * 16 + l16];
        acc[j] = (v8f){bv, bv, bv, bv, bv, bv, bv, bv};
    }
    const int node0 = (mt * 16 + l16) * 2;               // left sibling
    const v2f* a0 = (const v2f*)h + node0 * (NSTR / 2) + lhalf;
    const v2f* a1 = a0 + (NSTR / 2);                     // right sibling
#pragma unroll 8
    for (int ks = 0; ks < KS2 / 2; ++ks) {               // K = 0..255
        const v2f a = a0[ks * 2];                        // ds_load_b64
#pragma unroll
        for (int j = 0; j < 2; ++j) {
            const v2f bf = w2[(ks * 16 + ntp + j) * 32]; // global b64 + imm
            acc[j] = wmma_f32(a, bf, acc[j]);
        }
    }
#pragma unroll 8
    for (int ks = 0; ks < KS2 / 2; ++ks) {               // K = 256..511
        const v2f a = a1[ks * 2];
#pragma unroll
        for (int j = 0; j < 2; ++j) {
            const v2f bf = w2[((ks + KS2 / 2) * 16 + ntp + j) * 32];
            acc[j] = wmma_f32(a, bf, acc[j]);
        }
    }
}

__device__ __forceinline__ void store_unit(float* __restrict__ h, const v8f acc[2],
                                           int mt, int ntp, int l16, int lhalf,
                                           int Mtot) {
#pragma unroll
    for (int j = 0; j < 2; ++j) {
        const int ncol = (ntp + j) * 16 + l16;
#pragma unroll
        for (int v = 0; v < 8; ++v) {
            const int m = mt * 16 + v + 8 * lhalf;       // C/D: M = v + 8*(l/16)
            if (m < Mtot) h[m * NSTR + ncol] = acc[j][v];
        }
    }
}

// One tree level, in place: `rows` node vectors -> rows/2.  Units = 1 M-tile x
// 2 N-tiles; <=2 units per wave, buffered in registers across the barrier.
__device__ __forceinline__ void reduce_step(float* __restrict__ h,
                                            const float* __restrict__ W2p,
                                            const float* __restrict__ b2,
                                            int rows, int wave, int lane) {
    const int lhalf = lane >> 4, l16 = lane & 15;
    const int Mtot   = rows >> 1;
    const int Mtiles = (Mtot + 15) >> 4;
    const int units  = Mtiles * 8;
    const v2f* w2 = (const v2f*)W2p + lane;
    const int u0 = wave, u1 = wave + NWAVES;
    v8f acc0[2], acc1[2];
    if (u0 < units) tree_unit(h, w2, b2, u0 >> 3, (u0 & 7) * 2, l16, lhalf, acc0);
    if (u1 < units) tree_unit(h, w2, b2, u1 >> 3, (u1 & 7) * 2, l16, lhalf, acc1);
    __syncthreads();   // all reads of this level's inputs complete
    if (u0 < units) store_unit(h, acc0, u0 >> 3, (u0 & 7) * 2, l16, lhalf, Mtot);
    if (u1 < units) store_unit(h, acc1, u1 >> 3, (u1 & 7) * 2, l16, lhalf, Mtot);
    __syncthreads();   // level outputs visible
}

// Kernel 1: 128-leaf subtree per workgroup. Gathered-embedding GEMM
// (128x300 @ 300x256) into LDS, tree levels M=64/32/16 (no padding), emit 16
// nodes per workgroup.
__global__ __launch_bounds__(BLOCK) void rnn_leaf_subtree(
    const int*   __restrict__ word_ids,   // [BATCH][LEAVES]
    const float* __restrict__ emb,        // [VOCAB][WDIM]
    const float* __restrict__ W1p,        // packed [KS1*1024]
    const float* __restrict__ b1,
    const float* __restrict__ W2p,        // packed [KS2*1024]
    const float* __restrict__ b2,
    float*       __restrict__ nodes)      // [BATCH][NODES1][HID]
{
    extern __shared__ float h[];          // [SUBL][NSTR]
    __shared__ int wid[SUBL];

    const int tid   = threadIdx.x;
    const int wave  = tid >> 5;
    const int lane  = tid & 31;
    const int lhalf = lane >> 4;
    const int l16   = lane & 15;

    const int b = blockIdx.x >> 3;        // tree
    const int g = blockIdx.x & 7;         // 128-leaf block within tree

    if (tid < SUBL) wid[tid] = word_ids[b * LEAVES + g * SUBL + tid];
    __syncthreads();

    // ---- Leaf GEMM: 32 units of 2 M-tiles x 2 N-tiles (4 wmma chains) ----
    {
        const v2f* w1 = (const v2f*)W1p + lane;
#pragma unroll
        for (int it = 0; it < 2; ++it) {
            const int u   = wave + it * NWAVES;      // 0..31
            const int mt0 = (u >> 3) * 2;            // 0,2,4,6
            const int ntp = (u & 7) * 2;             // 0..14
            const v2f* e0 =
                (const v2f*)(emb + (long)wid[mt0 * 16 + l16]       * WDIM) + lhalf;
            const v2f* e1 =
                (const v2f*)(emb + (long)wid[(mt0 + 1) * 16 + l16] * WDIM) + lhalf;
            v8f acc[2][2];
#pragma unroll
            for (int j = 0; j < 2; ++j) {
                const float bv = b1[(ntp + j) * 16 + l16];
                acc[0][j] = (v8f){bv, bv, bv, bv, bv, bv, bv, bv};
                acc[1][j] = acc[0][j];
            }
#pragma unroll 5
            for (int ks = 0; ks < KS1; ++ks) {
                const v2f a0 = e0[ks * 2];                       // global b64
                const v2f a1 = e1[ks * 2];
#pragma unroll
                for (int j = 0; j < 2; ++j) {
                    const v2f bf = w1[(ks * 16 + ntp + j) * 32]; // b64 + imm
                    acc[0][j] = wmma_f32(a0, bf, acc[0][j]);
                    acc[1][j] = wmma_f32(a1, bf, acc[1][j]);
                }
            }
#pragma unroll
            for (int mi = 0; mi < 2; ++mi)
#pragma unroll
                for (int j = 0; j < 2; ++j) {
                    const int ncol = (ntp + j) * 16 + l16;
#pragma unroll
                    for (int v = 0; v < 8; ++v) {
                        const int m = (mt0 + mi) * 16 + v + 8 * lhalf;
                        h[m * NSTR + ncol] = acc[mi][j][v];
                    }
                }
        }
    }
    __syncthreads();

    // ---- Tree levels 128->64->32->16 nodes, all full 16-row tiles ----
    for (int rows = SUBL; rows > 16; rows >>= 1)
        reduce_step(h, W2p, b2, rows, wave, lane);

    // ---- Emit 16 node vectors (strip the stride padding) ----
    v2f* dst = (v2f*)(nodes + ((long)b * NODES1 + g * 16) * HID);
    const v2f* hs = (const v2f*)h;
    for (int i = tid; i < 16 * (HID / 2); i += BLOCK)
        dst[i] = hs[(i >> 7) * (NSTR / 2) + (i & 127)];
}

// Kernel 2: one workgroup per tree: 128 nodes -> root (7 levels).
__global__ __launch_bounds__(BLOCK) void rnn_top(
    const float* __restrict__ nodes,      // [BATCH][NODES1][HID]
    const float* __restrict__ W2p,
    const float* __restrict__ b2,
    float*       __restrict__ out)        // [BATCH][HID]
{
    extern __shared__ float h[];          // [SUBL][NSTR]
    const int tid  = threadIdx.x;
    const int wave = tid >> 5;
    const int lane = tid & 31;
    const int b    = blockIdx.x;

    const v2f* src = (const v2f*)(nodes + (long)b * NODES1 * HID);
    v2f* hd = (v2f*)h;
    for (int i = tid; i < NODES1 * (HID / 2); i += BLOCK)
        hd[(i >> 7) * (NSTR / 2) + (i & 127)] = src[i];
    __syncthreads();

    for (int rows = NODES1; rows > 1; rows >>= 1)
        reduce_step(h, W2p, b2, rows, wave, lane);

    if (tid < HID) out[b * HID + tid] = h[tid];
}

extern "C" void kernel_launch(void* const* d_in, const int* in_sizes, int n_in,
                              void* d_out, int out_size, void* d_ws, size_t ws_size,
                              hipStream_t stream) {
    const int*   word_ids = (const int*)  d_in[0];
    const float* emb      = (const float*)d_in[1];
    const float* W1       = (const float*)d_in[2];
    const float* b1       = (const float*)d_in[3];
    const float* W2       = (const float*)d_in[4];
    const float* b2       = (const float*)d_in[5];
    float*       out      = (float*)d_out;

    // Workspace (f32): nodes (33.5 MB) | W1p (300 KB) | W2p (512 KB)
    float* nodes = (float*)d_ws;
    float* W1p   = nodes + (size_t)BATCH * NODES1 * HID;
    float* W2p   = W1p + (size_t)KS1 * 1024;

    pack_weights<<<dim3((KS1 * 1024 + 255) / 256), dim3(256), 0, stream>>>(
        W1, W1p, KS1);
    pack_weights<<<dim3((KS2 * 1024 + 255) / 256), dim3(256), 0, stream>>>(
        W2, W2p, KS2);
    rnn_leaf_subtree<<<dim3(BATCH * (LEAVES / SUBL)), dim3(BLOCK), LDS_BYTES,
                       stream>>>(word_ids, emb, W1p, b1, W2p, b2, nodes);
    rnn_top<<<dim3(BATCH), dim3(BLOCK), LDS_BYTES, stream>>>(nodes, W2p, b2, out);
}